// TimemixingFC_channel_80865644249622
// MI455X (gfx1250) — compile-verified
//
#include <hip/hip_runtime.h>

typedef float v2f __attribute__((ext_vector_type(2)));
typedef float v8f __attribute__((ext_vector_type(8)));

#define C_  96
#define C3  288
#define D_  16
#define HW  1024
#define XP  100   // lds_x row pitch (floats) -> conflict-free column reads
#define QP  292   // lds_qkv row pitch (floats)

// Pack weight src[n][k] (n<N rows, k<K cols) into k-pair-interleaved B layout:
//   dst[(k>>1)*(2N) + 2n + (k&1)] = src[n*K + k]
// so a lane's (k, k+1) pair at column n is one aligned 8-byte word, contiguous in n.
__global__ void pack_b_k(const float* __restrict__ src, float* __restrict__ dst,
                         int N, int K) {
    int n_elems = N * K;
    for (int i = blockIdx.x * blockDim.x + threadIdx.x; i < n_elems;
         i += gridDim.x * blockDim.x) {
        int n = i / K, k = i % K;
        dst[(k >> 1) * (2 * N) + 2 * n + (k & 1)] = src[i];
    }
}

// One fused pass: QKV fc -> windowed "attention" (via G = k^T v trick) -> proj.
// `shift` folds jnp.roll into the temporal index for both read and write.
__global__ __launch_bounds__(256)
void fused_pass_kernel(const float* __restrict__ xin,
                       const float* __restrict__ wP,   // packed QKV weights, B[k][n], N=288
                       const float* __restrict__ qb,   // [288]
                       const float* __restrict__ pP,   // packed proj weights, B[k][n], N=96
                       const float* __restrict__ pb,   // [96]
                       float* __restrict__ out,
                       int shift)
{
    __shared__ float lds_x[32 * XP];     // X tile, later reused as O tile
    __shared__ float lds_qkv[32 * QP];   // QKV tile, later reused as OUT staging
    __shared__ float lds_g[16 * 4];      // per-window 2x2 G

    const int tid  = threadIdx.x;
    const int lane = tid & 31;
    const int wv   = tid >> 5;
    const int lh   = lane & 15;
    const int khi  = (lane >> 4) << 1;   // 0 for lanes 0-15, 2 for lanes 16-31

    // block -> (b, t, spatial base): 1024 blocks = 2 * 8 * 64
    const int bt = blockIdx.x >> 6;
    const int sp = (blockIdx.x & 63) << 4;
    const int b  = bt >> 3;
    const int t  = bt & 7;
    const int d0 = (2 * t     - shift + D_) % D_;
    const int d1 = (2 * t + 1 - shift + D_) % D_;

    const float* base0 = xin + ((size_t)(b * D_ + d0) * HW + sp) * C_;
    const float* base1 = xin + ((size_t)(b * D_ + d1) * HW + sp) * C_;

    // ---- stage 1: coalesced load X tile [32 rows x 96 ch] (rows 0-15: d0, 16-31: d1)
    for (int e4 = tid; e4 < 32 * 24; e4 += 256) {
        int row = e4 / 24;
        int c4  = (e4 % 24) * 4;
        const float* src = (row < 16) ? (base0 + (size_t)row * C_)
                                      : (base1 + (size_t)(row - 16) * C_);
        float4 v = *(const float4*)(src + c4);
        float* d = &lds_x[row * XP + c4];
        d[0] = v.x; d[1] = v.y; d[2] = v.z; d[3] = v.w;
    }
    __syncthreads();

    // ---- stage 2: QKV[32][288] = X @ W^T + qb
    // Each wave owns one 16-wide N tile and computes BOTH 16-row M tiles,
    // sharing each B fragment (1 x b64 load) across two WMMAs.
    for (int nt = wv; nt < 18; nt += 8) {
        int nbase = nt * 16;
        float bias = qb[nbase + lh];
        v8f acc0, acc1;
        #pragma unroll
        for (int i = 0; i < 8; ++i) { acc0[i] = bias; acc1[i] = bias; }
        const float* bcol = wP + ((khi >> 1) * (2 * C3) + 2 * (nbase + lh));
        #pragma unroll
        for (int k = 0; k < C_; k += 4) {
            v2f bf = *(const v2f*)(bcol + (k >> 1) * (2 * C3));
            v2f a0 = *(const v2f*)&lds_x[(lh)      * XP + k + khi];
            v2f a1 = *(const v2f*)&lds_x[(16 + lh) * XP + k + khi];
            acc0 = __builtin_amdgcn_wmma_f32_16x16x4_f32(false, a0, false, bf,
                                                         (short)0, acc0, false, false);
            acc1 = __builtin_amdgcn_wmma_f32_16x16x4_f32(false, a1, false, bf,
                                                         (short)0, acc1, false, false);
        }
        #pragma unroll
        for (int i = 0; i < 8; ++i) {
            int r = (lane < 16) ? i : i + 8;
            lds_qkv[r        * QP + nbase + lh] = acc0[i];
            lds_qkv[(16 + r) * QP + nbase + lh] = acc1[i];
        }
    }
    __syncthreads();

    // ---- stage 3: attention.  (q k^T) v  ==  q (k^T v);  G = k^T v is 2x2 per window.
    {
        int grp = tid >> 4;   // window pair 0..15
        int j   = tid & 15;
        float g00 = 0.f, g01 = 0.f, g10 = 0.f, g11 = 0.f;
        #pragma unroll
        for (int jj = 0; jj < 3; ++jj) {
            int cg = j + jj * 16;
            #pragma unroll
            for (int wn = 0; wn < 2; ++wn) {
                const float* r = &lds_qkv[(grp + wn * 16) * QP];
                float k0 = r[C_     + 2 * cg], k1 = r[C_     + 2 * cg + 1];
                float v0 = r[2 * C_ + 2 * cg], v1 = r[2 * C_ + 2 * cg + 1];
                g00 += k0 * v0; g01 += k0 * v1; g10 += k1 * v0; g11 += k1 * v1;
            }
        }
        #pragma unroll
        for (int m = 8; m >= 1; m >>= 1) {
            g00 += __shfl_xor(g00, m, 16);
            g01 += __shfl_xor(g01, m, 16);
            g10 += __shfl_xor(g10, m, 16);
            g11 += __shfl_xor(g11, m, 16);
        }
        if (j == 0) {
            lds_g[grp * 4 + 0] = g00; lds_g[grp * 4 + 1] = g01;
            lds_g[grp * 4 + 2] = g10; lds_g[grp * 4 + 3] = g11;
        }
    }
    __syncthreads();

    // O[m, 2cg+s] = q0*G[0][s] + q1*G[1][s]  -> overwrite lds_x
    {
        int grp = tid >> 4;
        int j   = tid & 15;
        float g00 = lds_g[grp * 4 + 0], g01 = lds_g[grp * 4 + 1];
        float g10 = lds_g[grp * 4 + 2], g11 = lds_g[grp * 4 + 3];
        #pragma unroll
        for (int wn = 0; wn < 2; ++wn) {
            int row = grp + wn * 16;
            #pragma unroll
            for (int jj = 0; jj < 3; ++jj) {
                int cg = j + jj * 16;
                float q0 = lds_qkv[row * QP + 2 * cg];
                float q1 = lds_qkv[row * QP + 2 * cg + 1];
                lds_x[row * XP + 2 * cg]     = q0 * g00 + q1 * g10;
                lds_x[row * XP + 2 * cg + 1] = q0 * g01 + q1 * g11;
            }
        }
    }
    __syncthreads();

    // ---- stage 4: OUT[32][96] = O @ P^T + pb (6 N-tile tasks, both M tiles per wave)
    for (int nt = wv; nt < 6; nt += 8) {
        int nbase = nt * 16;
        float bias = pb[nbase + lh];
        v8f acc0, acc1;
        #pragma unroll
        for (int i = 0; i < 8; ++i) { acc0[i] = bias; acc1[i] = bias; }
        const float* bcol = pP + ((khi >> 1) * (2 * C_) + 2 * (nbase + lh));
        #pragma unroll
        for (int k = 0; k < C_; k += 4) {
            v2f bf = *(const v2f*)(bcol + (k >> 1) * (2 * C_));
            v2f a0 = *(const v2f*)&lds_x[(lh)      * XP + k + khi];
            v2f a1 = *(const v2f*)&lds_x[(16 + lh) * XP + k + khi];
            acc0 = __builtin_amdgcn_wmma_f32_16x16x4_f32(false, a0, false, bf,
                                                         (short)0, acc0, false, false);
            acc1 = __builtin_amdgcn_wmma_f32_16x16x4_f32(false, a1, false, bf,
                                                         (short)0, acc1, false, false);
        }
        #pragma unroll
        for (int i = 0; i < 8; ++i) {
            int r = (lane < 16) ? i : i + 8;
            lds_qkv[r        * C_ + nbase + lh] = acc0[i];   // flat [32*96] staging
            lds_qkv[(16 + r) * C_ + nbase + lh] = acc1[i];
        }
    }
    __syncthreads();

    // ---- coalesced store (write back to the SAME frames we read: roll folded in)
    float* ob0 = out + ((size_t)(b * D_ + d0) * HW + sp) * C_;
    float* ob1 = out + ((size_t)(b * D_ + d1) * HW + sp) * C_;
    for (int e4 = tid; e4 < 32 * 24; e4 += 256) {
        int row = e4 / 24;
        int c4  = (e4 % 24) * 4;
        float* dst = (row < 16) ? (ob0 + (size_t)row * C_)
                                : (ob1 + (size_t)(row - 16) * C_);
        const float* s = &lds_qkv[row * C_ + c4];
        float4 v; v.x = s[0]; v.y = s[1]; v.z = s[2]; v.w = s[3];
        *(float4*)(dst + c4) = v;
    }
}

extern "C" void kernel_launch(void* const* d_in, const int* in_sizes, int n_in,
                              void* d_out, int out_size, void* d_ws, size_t ws_size,
                              hipStream_t stream) {
    const float* x       = (const float*)d_in[0];
    const float* lr_w    = (const float*)d_in[1];
    const float* lr_b    = (const float*)d_in[2];
    const float* proj_w  = (const float*)d_in[3];
    const float* proj_b  = (const float*)d_in[4];
    const float* sw_w    = (const float*)d_in[5];
    const float* sw_b    = (const float*)d_in[6];
    const float* projs_w = (const float*)d_in[7];
    const float* projs_b = (const float*)d_in[8];
    float* outp = (float*)d_out;

    float* ws  = (float*)d_ws;
    float* wP1 = ws;                       // 96*288 packed
    float* pP1 = wP1 + C_ * C3;            // 96*96 packed
    float* wP2 = pP1 + C_ * C_;            // 96*288 packed
    float* pP2 = wP2 + C_ * C3;            // 96*96 packed
    float* mid = pP2 + C_ * C_;            // B*D*H*W*C intermediate (pass-1 output)

    pack_b_k<<<64, 256, 0, stream>>>(lr_w,    wP1, C3, C_);
    pack_b_k<<<64, 256, 0, stream>>>(proj_w,  pP1, C_, C_);
    pack_b_k<<<64, 256, 0, stream>>>(sw_w,    wP2, C3, C_);
    pack_b_k<<<64, 256, 0, stream>>>(projs_w, pP2, C_, C_);

    dim3 grid(1024);  // B * (D/2) * (H*W/16)
    fused_pass_kernel<<<grid, 256, 0, stream>>>(x,   wP1, lr_b, pP1, proj_b,  mid,  0);
    fused_pass_kernel<<<grid, 256, 0, stream>>>(mid, wP2, sw_b, pP2, projs_b, outp, 1);
}